// InteractionBlock_24197845746071
// MI455X (gfx1250) — compile-verified
//
#include <hip/hip_runtime.h>

typedef float v2f __attribute__((ext_vector_type(2)));
typedef float v8f __attribute__((ext_vector_type(8)));

#define NNODES 20000
#define NEDGES 640000
#define FDIM   128
#define RBFD   64
#define TTYPES 5
#define DHEAD  16

__device__ __forceinline__ v8f wmma4(v2f a, v2f b, v8f c) {
  // V_WMMA_F32_16X16X4_F32: D = A(16x4) * B(4x16) + C(16x16), fp32 native
  return __builtin_amdgcn_wmma_f32_16x16x4_f32(false, a, false, b, (short)0, c, false, false);
}

__device__ __forceinline__ float silu_f(float x) { return x / (1.0f + __expf(-x)); }

// ---------------------------------------------------------------------------
// K1: xl = x @ lin1_w^T   [N,128]x[128,128]
// ---------------------------------------------------------------------------
__global__ __launch_bounds__(256) void k_xl(const float* __restrict__ x,
                                            const float* __restrict__ w,
                                            float* __restrict__ xl) {
  __shared__ float lw[FDIM * FDIM];
  for (int i = threadIdx.x; i < FDIM * FDIM; i += 256) lw[i] = w[i];
  __syncthreads();

  const int lane = threadIdx.x & 31;
  const int wv   = threadIdx.x >> 5;
  const int l16  = lane & 15;
  const int lhi  = lane >> 4;

  const int nTiles = NNODES / 16;
  const int gw = blockIdx.x * 8 + wv;
  const int gs = gridDim.x * 8;

  for (int t = gw; t < nTiles; t += gs) {
    const int row0 = t * 16;
    v2f afr[32];
    const v2f* ap = reinterpret_cast<const v2f*>(x + (row0 + l16) * FDIM + 2 * lhi);
#pragma unroll
    for (int kk = 0; kk < 32; ++kk) afr[kk] = ap[kk * 2];

    for (int ct = 0; ct < 8; ++ct) {
      v8f acc = {0, 0, 0, 0, 0, 0, 0, 0};
      const v2f* bp = reinterpret_cast<const v2f*>(&lw[(ct * 16 + l16) * FDIM + 2 * lhi]);
#pragma unroll
      for (int kk = 0; kk < 32; ++kk) acc = wmma4(afr[kk], bp[kk * 2], acc);
      float* op = xl + row0 * FDIM + ct * 16 + l16;
#pragma unroll
      for (int v = 0; v < 8; ++v) op[(v + 8 * lhi) * FDIM] = acc[v];
    }
  }
}

// ---------------------------------------------------------------------------
// K2: fused edge filter net + cutoff + message + scatter into y / cnt
// ---------------------------------------------------------------------------
__global__ __launch_bounds__(256) void k_edge(
    const float* __restrict__ edge_attr, const float* __restrict__ edge_weight,
    const int* __restrict__ eidx, const int* __restrict__ z,
    const float* __restrict__ fw1, const float* __restrict__ fb1,
    const float* __restrict__ fw2, const float* __restrict__ fb2,
    const float* __restrict__ xl, float* __restrict__ y, float* __restrict__ cnt) {
  // LDS: f_w2 (64KB) + per-wave h tile (8KB x 8) = 128KB
  __shared__ float lds[FDIM * FDIM + 8 * 16 * FDIM];
  float* lw2 = lds;
  for (int i = threadIdx.x; i < FDIM * FDIM; i += 256) lw2[i] = fw2[i];
  __syncthreads();

  const int lane = threadIdx.x & 31;
  const int wv   = threadIdx.x >> 5;
  const int l16  = lane & 15;
  const int lhi  = lane >> 4;
  float* hme = lds + FDIM * FDIM + wv * (16 * FDIM);

  const float PI_OVER_CUT = 0.62831853071795864769f;  // pi / 5
  const int nTiles = NEDGES / 16;
  const int gw = blockIdx.x * 8 + wv;
  const int gs = gridDim.x * 8;

  for (int t = gw; t < nTiles; t += gs) {
    const int e0 = t * 16;

    // per-row metadata held by lanes 0..15
    float cfl = 0.0f;
    int sl = 0, segl = 0;
    if (lane < 16) {
      const int e = e0 + lane;
      const float wgt = edge_weight[e];
      float c = 0.5f * (__cosf(wgt * PI_OVER_CUT) + 1.0f);
      cfl = (wgt < 5.0f) ? c : 0.0f;
      sl = eidx[e];
      const int d = eidx[NEDGES + e];
      segl = d * TTYPES + z[sl];
      atomicAdd(&cnt[segl], 1.0f);
    }

    // GEMM1: h = silu(edge_attr[16x64] @ fw1^T + b1)  (B frags from global; L2/WGP$ hot)
    v2f afr[32];
    const v2f* ap = reinterpret_cast<const v2f*>(edge_attr + (e0 + l16) * RBFD + 2 * lhi);
#pragma unroll
    for (int kk = 0; kk < 16; ++kk) afr[kk] = ap[kk * 2];

    for (int ct = 0; ct < 8; ++ct) {
      v8f acc = {0, 0, 0, 0, 0, 0, 0, 0};
      const v2f* bp = reinterpret_cast<const v2f*>(fw1 + (ct * 16 + l16) * RBFD + 2 * lhi);
#pragma unroll
      for (int kk = 0; kk < 16; ++kk) acc = wmma4(afr[kk], bp[kk * 2], acc);
      const int col = ct * 16 + l16;
      const float b1 = fb1[col];
#pragma unroll
      for (int v = 0; v < 8; ++v)
        hme[(v + 8 * lhi) * FDIM + col] = silu_f(acc[v] + b1);
    }

    // GEMM2: Wf = h @ fw2^T + b2, then msg = Wf * C * xl[src], scatter-add
    const v2f* a2p = reinterpret_cast<const v2f*>(hme + l16 * FDIM + 2 * lhi);
#pragma unroll
    for (int kk = 0; kk < 32; ++kk) afr[kk] = a2p[kk * 2];

    for (int ct = 0; ct < 8; ++ct) {
      v8f acc = {0, 0, 0, 0, 0, 0, 0, 0};
      const v2f* bp = reinterpret_cast<const v2f*>(&lw2[(ct * 16 + l16) * FDIM + 2 * lhi]);
#pragma unroll
      for (int kk = 0; kk < 32; ++kk) acc = wmma4(afr[kk], bp[kk * 2], acc);
      const int col = ct * 16 + l16;
      const float b2 = fb2[col];
#pragma unroll
      for (int v = 0; v < 8; ++v) {
        const int r   = v + 8 * lhi;
        const float c = __shfl(cfl, r, 32);
        const int s   = __shfl(sl, r, 32);
        const int seg = __shfl(segl, r, 32);
        const float m = (acc[v] + b2) * c * xl[s * FDIM + col];
        atomicAdd(&y[seg * FDIM + col], m);
      }
    }
  }
}

// ---------------------------------------------------------------------------
// K3: fused q/k/v GEMMs (WMMA, into LDS) + per-node masked SiLU attention
//     emits g[n,:] = sum_i sum_j attn[h,i,j]*v[j,:] and cntp[n] = sum_t pm
// ---------------------------------------------------------------------------
__global__ __launch_bounds__(256) void k_attn(
    const float* __restrict__ y, const float* __restrict__ cnt,
    const float* __restrict__ qw, const float* __restrict__ qb,
    const float* __restrict__ kw, const float* __restrict__ kb,
    const float* __restrict__ vw, const float* __restrict__ vb,
    float* __restrict__ g, float* __restrict__ cntp) {
  __shared__ float sQKV[3 * 80 * FDIM];  // q|k|v for 16 nodes x 5 slots (120KB)
  __shared__ float sAttn[16 * 200];      // 8 heads x 5 x 5 per node

  const int lane = threadIdx.x & 31;
  const int wv   = threadIdx.x >> 5;
  const int l16  = lane & 15;
  const int lhi  = lane >> 4;
  const int node0 = blockIdx.x * 16;

  const float* Ws[3] = {qw, kw, vw};
  const float* Bs[3] = {qb, kb, vb};

  // Phase A: 15 GEMM tile-tasks (3 matrices x 5 row-tiles of 16) over 8 waves
  for (int t = wv; t < 15; t += 8) {
    const int m = t / 5, rt = t % 5;
    const int row0 = node0 * TTYPES + rt * 16;
    const float* W = Ws[m];
    const float* B = Bs[m];

    v2f afr[32];
    const v2f* ap = reinterpret_cast<const v2f*>(y + (row0 + l16) * FDIM + 2 * lhi);
#pragma unroll
    for (int kk = 0; kk < 32; ++kk) afr[kk] = ap[kk * 2];

    float* dstl = sQKV + m * (80 * FDIM) + rt * 16 * FDIM;
    for (int ct = 0; ct < 8; ++ct) {
      v8f acc = {0, 0, 0, 0, 0, 0, 0, 0};
      const v2f* bp = reinterpret_cast<const v2f*>(W + (ct * 16 + l16) * FDIM + 2 * lhi);
#pragma unroll
      for (int kk = 0; kk < 32; ++kk) acc = wmma4(afr[kk], bp[kk * 2], acc);
      const int col = ct * 16 + l16;
      const float bias = B[col];
#pragma unroll
      for (int v = 0; v < 8; ++v)
        dstl[(v + 8 * lhi) * FDIM + col] = acc[v] + bias;
    }
  }
  __syncthreads();

  // Phase B: attention, 2 nodes per wave
  for (int nn = wv; nn < 16; nn += 8) {
    const int node = node0 + nn;
    const int lr = nn * 5;
    float pm[TTYPES];
#pragma unroll
    for (int j = 0; j < TTYPES; ++j)
      pm[j] = (cnt[node * TTYPES + j] > 0.0f) ? 1.0f : 0.0f;

    // attn[h,i,j] = silu(q_i . k_j) * pm_i * pm_j  (200 scalars, lanes cooperate)
    for (int p = lane; p < 200; p += 32) {
      const int h = p / 25, ij = p % 25, i = ij / 5, j = ij % 5;
      const float* qr = sQKV + (lr + i) * FDIM + h * DHEAD;
      const float* kr = sQKV + 80 * FDIM + (lr + j) * FDIM + h * DHEAD;
      float dot = 0.0f;
#pragma unroll
      for (int d = 0; d < DHEAD; ++d) dot += qr[d] * kr[d];
      sAttn[nn * 200 + p] = silu_f(dot) * pm[i] * pm[j];
    }

    // g[col] = sum_j (sum_i attn[h,i,j]) * v[j,col]; 4 cols per lane
    const int col = lane * 4;
    const int h = col >> 4;
    float g0 = 0, g1 = 0, g2 = 0, g3 = 0;
#pragma unroll
    for (int j = 0; j < TTYPES; ++j) {
      float cs = 0.0f;
#pragma unroll
      for (int i = 0; i < TTYPES; ++i) cs += sAttn[nn * 200 + h * 25 + i * 5 + j];
      const float* vr = sQKV + 2 * (80 * FDIM) + (lr + j) * FDIM + col;
      g0 += cs * vr[0]; g1 += cs * vr[1]; g2 += cs * vr[2]; g3 += cs * vr[3];
    }
    float* go = g + node * FDIM + col;
    go[0] = g0; go[1] = g1; go[2] = g2; go[3] = g3;
    if (lane == 0) cntp[node] = pm[0] + pm[1] + pm[2] + pm[3] + pm[4];
  }
}

// ---------------------------------------------------------------------------
// K4: out = silu((g @ o1^T + cntp*o1_b) @ o2^T + o2_b)
// ---------------------------------------------------------------------------
__global__ __launch_bounds__(256) void k_out(
    const float* __restrict__ g, const float* __restrict__ cntp,
    const float* __restrict__ o1w, const float* __restrict__ o1b,
    const float* __restrict__ o2w, const float* __restrict__ o2b,
    float* __restrict__ out) {
  __shared__ float lds[FDIM * FDIM + 8 * 16 * FDIM];  // o1w + per-wave tmp tile (128KB)
  float* lw1 = lds;
  for (int i = threadIdx.x; i < FDIM * FDIM; i += 256) lw1[i] = o1w[i];
  __syncthreads();

  const int lane = threadIdx.x & 31;
  const int wv   = threadIdx.x >> 5;
  const int l16  = lane & 15;
  const int lhi  = lane >> 4;
  float* hme = lds + FDIM * FDIM + wv * (16 * FDIM);

  const int nTiles = NNODES / 16;
  const int gw = blockIdx.x * 8 + wv;
  const int gs = gridDim.x * 8;

  for (int t = gw; t < nTiles; t += gs) {
    const int row0 = t * 16;
    float cpl = (lane < 16) ? cntp[row0 + lane] : 0.0f;

    v2f afr[32];
    const v2f* ap = reinterpret_cast<const v2f*>(g + (row0 + l16) * FDIM + 2 * lhi);
#pragma unroll
    for (int kk = 0; kk < 32; ++kk) afr[kk] = ap[kk * 2];

    for (int ct = 0; ct < 8; ++ct) {
      v8f acc = {0, 0, 0, 0, 0, 0, 0, 0};
      const v2f* bp = reinterpret_cast<const v2f*>(&lw1[(ct * 16 + l16) * FDIM + 2 * lhi]);
#pragma unroll
      for (int kk = 0; kk < 32; ++kk) acc = wmma4(afr[kk], bp[kk * 2], acc);
      const int col = ct * 16 + l16;
      const float b1 = o1b[col];
#pragma unroll
      for (int v = 0; v < 8; ++v) {
        const float cp = __shfl(cpl, v + 8 * lhi, 32);
        hme[(v + 8 * lhi) * FDIM + col] = acc[v] + cp * b1;
      }
    }

    const v2f* a2p = reinterpret_cast<const v2f*>(hme + l16 * FDIM + 2 * lhi);
#pragma unroll
    for (int kk = 0; kk < 32; ++kk) afr[kk] = a2p[kk * 2];

    for (int ct = 0; ct < 8; ++ct) {
      v8f acc = {0, 0, 0, 0, 0, 0, 0, 0};
      const v2f* bp = reinterpret_cast<const v2f*>(o2w + (ct * 16 + l16) * FDIM + 2 * lhi);
#pragma unroll
      for (int kk = 0; kk < 32; ++kk) acc = wmma4(afr[kk], bp[kk * 2], acc);
      const int col = ct * 16 + l16;
      const float b2 = o2b[col];
#pragma unroll
      for (int v = 0; v < 8; ++v)
        out[(row0 + v + 8 * lhi) * FDIM + col] = silu_f(acc[v] + b2);
    }
  }
}

// ---------------------------------------------------------------------------
extern "C" void kernel_launch(void* const* d_in, const int* in_sizes, int n_in,
                              void* d_out, int out_size, void* d_ws, size_t ws_size,
                              hipStream_t stream) {
  const float* x     = (const float*)d_in[0];
  const int*   z     = (const int*)d_in[1];
  const int*   eidx  = (const int*)d_in[2];
  const float* ew    = (const float*)d_in[3];
  const float* ea    = (const float*)d_in[4];
  const float* lin1w = (const float*)d_in[5];
  const float* fw1   = (const float*)d_in[6];
  const float* fb1   = (const float*)d_in[7];
  const float* fw2   = (const float*)d_in[8];
  const float* fb2   = (const float*)d_in[9];
  const float* qw    = (const float*)d_in[10];
  const float* qb    = (const float*)d_in[11];
  const float* kw    = (const float*)d_in[12];
  const float* kb    = (const float*)d_in[13];
  const float* vw    = (const float*)d_in[14];
  const float* vb    = (const float*)d_in[15];
  const float* o1w   = (const float*)d_in[16];
  const float* o1b   = (const float*)d_in[17];
  const float* o2w   = (const float*)d_in[18];
  const float* o2b   = (const float*)d_in[19];

  float* ws  = (float*)d_ws;
  float* xl  = ws;                              // N*F
  float* y   = xl + NNODES * FDIM;              // N*T*F
  float* cnt = y + NNODES * TTYPES * FDIM;      // N*T   (contiguous with y)
  float* g   = cnt + NNODES * TTYPES;           // N*F
  float* cp  = g + NNODES * FDIM;               // N

  hipMemsetAsync(y, 0, sizeof(float) * (NNODES * TTYPES * FDIM + NNODES * TTYPES), stream);

  k_xl<<<64, 256, 0, stream>>>(x, lin1w, xl);
  k_edge<<<512, 256, 0, stream>>>(ea, ew, eidx, z, fw1, fb1, fw2, fb2, xl, y, cnt);
  k_attn<<<NNODES / 16, 256, 0, stream>>>(y, cnt, qw, qb, kw, kb, vw, vb, g, cp);
  k_out<<<128, 256, 0, stream>>>(g, cp, o1w, o1b, o2w, o2b, (float*)d_out);
}